// CoAttentionMessagePassingNetwork_9740985827685
// MI455X (gfx1250) — compile-verified
//
#include <hip/hip_runtime.h>
#include <math.h>

// ---------------------------------------------------------------------------
// CoAttention message-passing network for MI455X (gfx1250, wave32, WMMA).
// All dense 64-wide GEMMs use V_WMMA_F32_16X16X4_F32 (fp32 in, fp32 acc).
// Irregular gather/scatter phases use coalesced wave-per-edge streaming and
// hardware float atomics; segment-max uses the ordered-int atomic trick.
// ---------------------------------------------------------------------------

#define NN     32768
#define BB     1024
#define HH     64
#define RR     64
#define E_IN   131072
#define E_OUT  524288
#define INV_T  0.125f          // 1/sqrt(H)
#define SM_EPS 1.1920929e-7f   // float32 machine eps (reference softmax eps)
#define LN_EPS 1e-5f
#define SLOPE  0.01f

typedef __attribute__((ext_vector_type(2))) float v2f;
typedef __attribute__((ext_vector_type(8))) float v8f;

__device__ __forceinline__ v8f wmma4(v2f a, v2f b, v8f c) {
    // D = A(16x4) * B(4x16) + C(16x16), fp32
    return __builtin_amdgcn_wmma_f32_16x16x4_f32(
        false, a, false, b, (short)0, c, false, false);
}

__device__ __forceinline__ float lrelu(float v) {
    return v > 0.0f ? v : SLOPE * v;
}

__device__ __forceinline__ void atomicMaxF(float* addr, float v) {
    if (v >= 0.0f) atomicMax((int*)addr, __float_as_int(v));
    else           atomicMin((unsigned int*)addr, (unsigned int)__float_as_int(v));
}

// ---------------------------------------------------------------------------
// Y[rows x 64] = act( X[rows x 64] @ W[64 x 64]^T + bias )
// Block = 4 waves, one 16-row tile; wave w owns output cols [16w,16w+16).
// A-frag (16x4 fp32): lanes 0..15 -> rows, (K,K+1); lanes 16..31 -> (K+2,K+3).
// B-frag (4x16): W row-major means B[k][n] = W[n][k] -> same gather pattern.
// Safe for in-place (Y==X): all A loads complete before the barrier.
// ---------------------------------------------------------------------------
__global__ __launch_bounds__(128)
void lin64_kernel(const float* __restrict__ X, const float* __restrict__ W,
                  const float* __restrict__ bias, float* __restrict__ Y, int act)
{
    const int lane = threadIdx.x & 31;
    const int wv   = threadIdx.x >> 5;
    const int m0   = blockIdx.x * 16;
    const int n0   = wv * 16;
    const int l    = lane & 15;
    const int hi   = lane >> 4;

    const float* xrow = X + (size_t)(m0 + l) * HH + 2 * hi;
    const float* wrow = W + (size_t)(n0 + l) * HH + 2 * hi;

    v8f acc = {};
    #pragma unroll
    for (int kb = 0; kb < HH; kb += 4) {
        v2f a = *(const v2f*)(xrow + kb);
        v2f b = *(const v2f*)(wrow + kb);
        acc = wmma4(a, b, acc);
    }
    __syncthreads();                         // allow safe in-place operation
    const int col = n0 + l;
    const float bv = bias ? bias[col] : 0.0f;
    #pragma unroll
    for (int g = 0; g < 8; ++g) {
        const int row = m0 + g + hi * 8;     // C/D layout: vgpr g -> M=g (+8 hi)
        float v = acc[g] + bv;
        if (act) v = lrelu(v);
        Y[(size_t)row * HH + col] = v;
    }
}

// ---------------------------------------------------------------------------
// Fused edge MLP + compose + scatter (message_passing):
//   h2 = lrelu(lrelu(bond @ We1^T + be1) @ We2^T + be2)
//   out[seg[e]] += wn[idx[e]] * h2[e]        (atomic)
// One wave per 16-edge tile; h1 staged in LDS (stride 68 kills bank conflicts).
// ---------------------------------------------------------------------------
__global__ __launch_bounds__(128)
void edge_mp_kernel(const float* __restrict__ bond,
                    const int*   __restrict__ segid,
                    const int*   __restrict__ srcid,
                    const float* __restrict__ We1, const float* __restrict__ be1,
                    const float* __restrict__ We2, const float* __restrict__ be2,
                    const float* __restrict__ wn,   // atom @ Wn^T, [N,64]
                    float*       __restrict__ out)  // [N,64] pre-zeroed
{
    __shared__ float h1s[4][16 * 68];
    __shared__ int   sIdx[4][16];
    __shared__ int   sSeg[4][16];

    const int lane = threadIdx.x & 31;
    const int wv   = threadIdx.x >> 5;
    const int l    = lane & 15;
    const int hi   = lane >> 4;
    const int e0   = (blockIdx.x * 4 + wv) * 16;

    if (lane < 16) {
        sIdx[wv][lane] = srcid[e0 + lane];
        sSeg[wv][lane] = segid[e0 + lane];
    }

    // ---- GEMM1: h1 = lrelu(bond_tile @ We1^T + be1) ----
    {
        const float* xrow = bond + (size_t)(e0 + l) * HH + 2 * hi;
        v8f acc[4] = {};
        #pragma unroll
        for (int kb = 0; kb < HH; kb += 4) {
            v2f a = *(const v2f*)(xrow + kb);
            #pragma unroll
            for (int nt = 0; nt < 4; ++nt) {
                v2f b = *(const v2f*)(We1 + (size_t)(nt * 16 + l) * HH + 2 * hi + kb);
                acc[nt] = wmma4(a, b, acc[nt]);
            }
        }
        #pragma unroll
        for (int nt = 0; nt < 4; ++nt) {
            const int col = nt * 16 + l;
            const float bv = be1[col];
            #pragma unroll
            for (int g = 0; g < 8; ++g)
                h1s[wv][(g + hi * 8) * 68 + col] = lrelu(acc[nt][g] + bv);
        }
    }
    __syncthreads();

    // ---- GEMM2: h2 = lrelu(h1 @ We2^T + be2), A-frags from LDS ----
    v8f acc2[4] = {};
    {
        const float* arow = &h1s[wv][l * 68 + 2 * hi];
        #pragma unroll
        for (int kb = 0; kb < HH; kb += 4) {
            v2f a = *(const v2f*)(arow + kb);
            #pragma unroll
            for (int nt = 0; nt < 4; ++nt) {
                v2f b = *(const v2f*)(We2 + (size_t)(nt * 16 + l) * HH + 2 * hi + kb);
                acc2[nt] = wmma4(a, b, acc2[nt]);
            }
        }
    }

    // ---- compose with gathered node projection, scatter-add ----
    #pragma unroll
    for (int nt = 0; nt < 4; ++nt) {
        const int col = nt * 16 + l;
        const float bv = be2[col];
        #pragma unroll
        for (int g = 0; g < 8; ++g) {
            const int m   = g + hi * 8;
            const int src = sIdx[wv][m];
            const int dst = sSeg[wv][m];
            const float v = lrelu(acc2[nt][g] + bv) * wn[(size_t)src * HH + col];
            atomicAdd(&out[(size_t)dst * HH + col], v);
        }
    }
}

// ---------------------------------------------------------------------------
// t[e] = <kl[sl[e]], kr[sr[e]]>; atomic segment-max both sides.
// ---------------------------------------------------------------------------
__global__ __launch_bounds__(256)
void dotmax_kernel(const float* __restrict__ kl, const float* __restrict__ kr,
                   const int* __restrict__ sl, const int* __restrict__ sr,
                   float* __restrict__ t, float* __restrict__ maxl,
                   float* __restrict__ maxr)
{
    const int e = blockIdx.x * 256 + threadIdx.x;
    const int a = sl[e], b = sr[e];
    const float4* pa = (const float4*)(kl + (size_t)a * HH);
    const float4* pb = (const float4*)(kr + (size_t)b * HH);
    float s = 0.0f;
    #pragma unroll
    for (int i = 0; i < 16; ++i) {
        const float4 x = pa[i], y = pb[i];
        s += x.x * y.x + x.y * y.y + x.z * y.z + x.w * y.w;
    }
    t[e] = s;
    atomicMaxF(&maxl[a], s);
    atomicMaxF(&maxr[b], s);
}

__global__ __launch_bounds__(256)
void expnorm_kernel(const float* __restrict__ t,
                    const int* __restrict__ sl, const int* __restrict__ sr,
                    const float* __restrict__ maxl, const float* __restrict__ maxr,
                    float* __restrict__ el, float* __restrict__ er,
                    float* __restrict__ norml, float* __restrict__ normr)
{
    const int e = blockIdx.x * 256 + threadIdx.x;
    const int a = sl[e], b = sr[e];
    const float tv = t[e];
    const float x = __expf((tv - maxl[a]) * INV_T);
    const float y = __expf((tv - maxr[b]) * INV_T);
    el[e] = x; er[e] = y;
    atomicAdd(&norml[a], x);
    atomicAdd(&normr[b], y);
}

// ---------------------------------------------------------------------------
// Weighted scatter: ml[sl] += a_l * vr[sr]; mr[sr] += a_r * vl[sl].
// One wave per edge, 2 cols/lane.
// ---------------------------------------------------------------------------
__global__ __launch_bounds__(256)
void scatter_kernel(const float* __restrict__ vl, const float* __restrict__ vr,
                    const int* __restrict__ sl, const int* __restrict__ sr,
                    const float* __restrict__ el, const float* __restrict__ er,
                    const float* __restrict__ norml, const float* __restrict__ normr,
                    float* __restrict__ ml, float* __restrict__ mr)
{
    const int lane = threadIdx.x & 31;
    const int e    = blockIdx.x * 8 + (threadIdx.x >> 5);
    const int a = sl[e], b = sr[e];
    const float al = el[e] / (norml[a] + SM_EPS);
    const float ar = er[e] / (normr[b] + SM_EPS);
    const int c0 = lane, c1 = lane + 32;
    const size_t ra = (size_t)a * HH, rb = (size_t)b * HH;
    atomicAdd(&ml[ra + c0], al * vr[rb + c0]);
    atomicAdd(&ml[ra + c1], al * vr[rb + c1]);
    atomicAdd(&mr[rb + c0], ar * vl[ra + c0]);
    atomicAdd(&mr[rb + c1], ar * vl[ra + c1]);
}

// ---------------------------------------------------------------------------
// Fused residual + LayerNorm + per-molecule segment-sum (wave32 reductions).
// ---------------------------------------------------------------------------
__global__ __launch_bounds__(256)
void ln_molsum_kernel(const float* __restrict__ atom, const float* __restrict__ inn,
                      const float* __restrict__ outc,
                      const float* __restrict__ g, const float* __restrict__ bta,
                      const int* __restrict__ mol, float* __restrict__ msum)
{
    const int lane = threadIdx.x & 31;
    const int row  = blockIdx.x * 8 + (threadIdx.x >> 5);
    const size_t base = (size_t)row * HH;
    const int c0 = lane, c1 = lane + 32;
    float x0 = atom[base + c0] + inn[base + c0] + outc[base + c0];
    float x1 = atom[base + c1] + inn[base + c1] + outc[base + c1];
    float s = x0 + x1;
    #pragma unroll
    for (int m = 16; m >= 1; m >>= 1) s += __shfl_xor(s, m, 32);
    const float mu = s * (1.0f / 64.0f);
    const float d0 = x0 - mu, d1 = x1 - mu;
    float q = d0 * d0 + d1 * d1;
    #pragma unroll
    for (int m = 16; m >= 1; m >>= 1) q += __shfl_xor(q, m, 32);
    const float rstd = rsqrtf(q * (1.0f / 64.0f) + LN_EPS);
    const int mb = mol[row];
    atomicAdd(&msum[(size_t)mb * HH + c0], d0 * rstd * g[c0] + bta[c0]);
    atomicAdd(&msum[(size_t)mb * HH + c1], d1 * rstd * g[c1] + bta[c1]);
}

__global__ void fill_kernel(float* __restrict__ p, float v, int n) {
    const int i = blockIdx.x * 256 + threadIdx.x;
    if (i < n) p[i] = v;
}

// ---------------------------------------------------------------------------
extern "C" void kernel_launch(void* const* d_in, const int* in_sizes, int n_in,
                              void* d_out, int out_size, void* d_ws, size_t ws_size,
                              hipStream_t stream)
{
    (void)in_sizes; (void)n_in; (void)out_size; (void)ws_size;

    const int*   mol_l  = (const int*)  d_in[0];
    const float* atom_l = (const float*)d_in[1];
    const float* bond_l = (const float*)d_in[2];
    const int*   iseg_l = (const int*)  d_in[3];
    const int*   iidx_l = (const int*)  d_in[4];
    const int*   oseg_l = (const int*)  d_in[5];
    /* d_in[6] outer_index_left: redundant with segment ops */
    const int*   mol_r  = (const int*)  d_in[7];
    const float* atom_r = (const float*)d_in[8];
    const float* bond_r = (const float*)d_in[9];
    const int*   iseg_r = (const int*)  d_in[10];
    const int*   iidx_r = (const int*)  d_in[11];
    const int*   oseg_r = (const int*)  d_in[12];
    /* d_in[13] outer_index_right: unused */
    const float* Wk   = (const float*)d_in[14];
    const float* Wv   = (const float*)d_in[15];
    const float* Wco  = (const float*)d_in[16];
    const float* bco  = (const float*)d_in[17];
    const float* Wn   = (const float*)d_in[18];
    const float* We1  = (const float*)d_in[19];
    const float* be1  = (const float*)d_in[20];
    const float* We2  = (const float*)d_in[21];
    const float* be2  = (const float*)d_in[22];
    const float* ln_g = (const float*)d_in[23];
    const float* ln_b = (const float*)d_in[24];
    const float* Wr   = (const float*)d_in[25];
    const float* br   = (const float*)d_in[26];
    float* out = (float*)d_out;

    // Workspace layout (floats). bufA/bufB recycled: kl/kr -> ml/mr -> outl/outr.
    const size_t NHf = (size_t)NN * HH;
    float* ws    = (float*)d_ws;
    float* bufA  = ws;                       // kl, then m_l, then out_l (in-place)
    float* bufB  = bufA  + NHf;              // kr, then m_r, then out_r
    float* vl    = bufB  + NHf;
    float* vr    = vl    + NHf;
    float* wnl   = vr    + NHf;              // atom_l @ Wn^T
    float* wnr   = wnl   + NHf;
    float* inl   = wnr   + NHf;              // message-passing accumulators
    float* inr   = inl   + NHf;
    float* tt    = inr   + NHf;              // attention logits
    float* el    = tt    + E_OUT;
    float* er    = el    + E_OUT;
    float* maxl  = er    + E_OUT;            // segment maxima (ordered-int atomics)
    float* maxr  = maxl  + NN;
    float* norml = maxr  + NN;
    float* normr = norml + NN;
    float* msuml = normr + NN;               // per-molecule sums
    float* msumr = msuml + (size_t)BB * HH;

    const dim3 blk128(128), blk256(256);
    const int gRows  = NN / 16;              // lin64 on atoms
    const int gRead  = BB / 16;              // lin64 readout
    const int gEdgeO = E_OUT / 256;          // thread-per-edge passes
    const int gScat  = E_OUT / 8;            // wave-per-edge scatter
    const int gMP    = E_IN / 64;            // 4 waves x 16 edges per block
    const int gLN    = NN / 8;

    // --- init ---
    fill_kernel<<<(2 * NN + 255) / 256, blk256, 0, stream>>>(maxl, -INFINITY, 2 * NN);
    fill_kernel<<<(2 * NN + 2 * BB * HH + 255) / 256, blk256, 0, stream>>>(
        norml, 0.0f, 2 * NN + 2 * BB * HH);
    fill_kernel<<<(int)((2 * NHf + 255) / 256), blk256, 0, stream>>>(inl, 0.0f, (int)(2 * NHf));

    // --- dense projections (WMMA) ---
    lin64_kernel<<<gRows, blk128, 0, stream>>>(atom_l, Wk, nullptr, bufA, 0); // kl
    lin64_kernel<<<gRows, blk128, 0, stream>>>(atom_r, Wk, nullptr, bufB, 0); // kr
    lin64_kernel<<<gRows, blk128, 0, stream>>>(atom_l, Wv, nullptr, vl,   0);
    lin64_kernel<<<gRows, blk128, 0, stream>>>(atom_r, Wv, nullptr, vr,   0);
    lin64_kernel<<<gRows, blk128, 0, stream>>>(atom_l, Wn, nullptr, wnl,  0);
    lin64_kernel<<<gRows, blk128, 0, stream>>>(atom_r, Wn, nullptr, wnr,  0);

    // --- co-attention: logits + segment softmax + weighted scatter ---
    dotmax_kernel<<<gEdgeO, blk256, 0, stream>>>(bufA, bufB, oseg_l, oseg_r,
                                                 tt, maxl, maxr);
    expnorm_kernel<<<gEdgeO, blk256, 0, stream>>>(tt, oseg_l, oseg_r, maxl, maxr,
                                                  el, er, norml, normr);
    fill_kernel<<<(int)((2 * NHf + 255) / 256), blk256, 0, stream>>>(bufA, 0.0f,
                                                                     (int)(2 * NHf));
    scatter_kernel<<<gScat, blk256, 0, stream>>>(vl, vr, oseg_l, oseg_r,
                                                 el, er, norml, normr, bufA, bufB);
    lin64_kernel<<<gRows, blk128, 0, stream>>>(bufA, Wco, bco, bufA, 1); // out_l (in-place)
    lin64_kernel<<<gRows, blk128, 0, stream>>>(bufB, Wco, bco, bufB, 1); // out_r

    // --- intra-molecule message passing (fused edge MLP, WMMA) ---
    edge_mp_kernel<<<gMP, blk128, 0, stream>>>(bond_l, iseg_l, iidx_l,
                                               We1, be1, We2, be2, wnl, inl);
    edge_mp_kernel<<<gMP, blk128, 0, stream>>>(bond_r, iseg_r, iidx_r,
                                               We1, be1, We2, be2, wnr, inr);

    // --- residual + LayerNorm + molecule segment-sum ---
    ln_molsum_kernel<<<gLN, blk256, 0, stream>>>(atom_l, inl, bufA, ln_g, ln_b,
                                                 mol_l, msuml);
    ln_molsum_kernel<<<gLN, blk256, 0, stream>>>(atom_r, inr, bufB, ln_g, ln_b,
                                                 mol_r, msumr);

    // --- readout GEMMs straight into d_out (gl, then gr) ---
    lin64_kernel<<<gRead, blk128, 0, stream>>>(msuml, Wr, br, out, 0);
    lin64_kernel<<<gRead, blk128, 0, stream>>>(msumr, Wr, br, out + (size_t)BB * RR, 0);
}